// SimpleGraphEncoder_86543591014918
// MI455X (gfx1250) — compile-verified
//
#include <hip/hip_runtime.h>
#include <hip/hip_bf16.h>

typedef __attribute__((ext_vector_type(2))) float v2f;
typedef __attribute__((ext_vector_type(4))) float v4f;
typedef __attribute__((ext_vector_type(8))) float v8f;

#define HIDDEN 64
#define DNODE 384

// ---------------------------------------------------------------------------
// fp32 WMMA 16x16x4 wrapper: D = A(16x4) * B(4x16) + C, full fp32 precision.
// ---------------------------------------------------------------------------
static __device__ __forceinline__ v8f wmma4(v2f a, v2f b, v8f c) {
  return __builtin_amdgcn_wmma_f32_16x16x4_f32(
      /*neg_a=*/false, a, /*neg_b=*/false, b,
      /*c_mod=*/(short)0, c, /*reuse_a=*/false, /*reuse_b=*/false);
}

// Stage a [rows x 64] row-major weight block from global into LDS in
// K-pair-interleaved form: sW[(k/2)*128 + col*2 + (k&1)] = W[k][col].
// Each lane's WMMA B operand {W[kb][col], W[kb+1][col]} (kb even) then is one
// aligned 8-byte word -> single ds_load_b64, no register repacking.
static __device__ __forceinline__ void stage_pairs(float* sW, const float* W,
                                                   int rows, int tid, int nthr) {
  const int nvec = rows * HIDDEN / 4;
  for (int idx = tid; idx < nvec; idx += nthr) {
    int k = idx >> 4;            // row
    int col = (idx & 15) * 4;    // 4 consecutive cols
    v4f v = ((const v4f*)W)[idx];
    float* dstp = sW + (k >> 1) * 128 + (k & 1);
    dstp[(col + 0) * 2] = v.x;
    dstp[(col + 1) * 2] = v.y;
    dstp[(col + 2) * 2] = v.z;
    dstp[(col + 3) * 2] = v.w;
  }
}

// ---------------------------------------------------------------------------
// Generic float zero (grid-stride).
// ---------------------------------------------------------------------------
__global__ void k_zero(float* __restrict__ p, size_t n) {
  size_t i = (size_t)blockIdx.x * blockDim.x + threadIdx.x;
  size_t stride = (size_t)gridDim.x * blockDim.x;
  for (; i < n; i += stride) p[i] = 0.0f;
}

// ---------------------------------------------------------------------------
// In-degree: one thread per edge, f32 atomic into deg[dst].
// ---------------------------------------------------------------------------
__global__ void k_deg(const int* __restrict__ dst, float* __restrict__ deg, int E) {
  int e = blockIdx.x * blockDim.x + threadIdx.x;
  if (e < E) atomicAdd(deg + dst[e], 1.0f);
}

// ---------------------------------------------------------------------------
// Edge scatter: one wave per edge, 2 features per lane. h/agg are L2-resident
// (25.6 MB each, 192 MB L2) so this runs at L2/atomic throughput.
// ---------------------------------------------------------------------------
__global__ __launch_bounds__(256) void k_scatter(const int* __restrict__ src,
                                                 const int* __restrict__ dst,
                                                 const float* __restrict__ h,
                                                 float* __restrict__ agg, int E) {
  int t = blockIdx.x * blockDim.x + threadIdx.x;
  int e = t >> 5;
  int lane = t & 31;
  if (e >= E) return;
  int s = src[e];
  int d = dst[e];
  v2f v = *(const v2f*)(h + (size_t)s * HIDDEN + lane * 2);
  atomicAdd(agg + (size_t)d * HIDDEN + lane * 2, v.x);
  atomicAdd(agg + (size_t)d * HIDDEN + lane * 2 + 1, v.y);
}

// ---------------------------------------------------------------------------
// Input projection: h = relu(x @ W_in + b_in).  [N,384]x[384,64].
// One wave = 16 rows x 64 cols (4 WMMA accumulators). W_in staged into LDS
// in 96-row pair-interleaved chunks (24 KB). Memory-bound on x (153.6 MB).
// ---------------------------------------------------------------------------
__global__ __launch_bounds__(128) void k_inproj(const float* __restrict__ x,
                                                const float* __restrict__ Win,
                                                const float* __restrict__ bin,
                                                float* __restrict__ h, int n) {
  __shared__ float sW[96 * HIDDEN];  // 24 KB chunk (pair-interleaved)
  const int wave = threadIdx.x >> 5;
  const int lane = threadIdx.x & 31;
  const int hf = lane >> 4;   // half-wave select
  const int r = lane & 15;    // row (A) / col (B,C) within tile
  const int rowBase = blockIdx.x * 64 + wave * 16;
  const int row = rowBase + r;
  const int rowc = row < n ? row : n - 1;  // clamp loads; stores are guarded
  const float* xr = x + (size_t)rowc * DNODE;

  v8f acc[4];
#pragma unroll
  for (int c = 0; c < 4; ++c) {
    float b = bin[c * 16 + r];
#pragma unroll
    for (int i = 0; i < 8; ++i) acc[c][i] = b;
  }

  for (int kc = 0; kc < DNODE; kc += 96) {
    __syncthreads();
    stage_pairs(sW, Win + (size_t)kc * HIDDEN, 96, threadIdx.x, 128);
    __syncthreads();

    // prefetch next chunk of this row of x into cache while we compute
    if (kc + 96 < DNODE) __builtin_prefetch(xr + kc + 96 + 2 * hf, 0, 3);

#pragma unroll 4
    for (int k = 0; k < 96; k += 4) {
      // A: lane r holds row, K = k + 2*hf + {0,1}
      v2f a = *(const v2f*)(xr + kc + k + 2 * hf);
      const float* wrow = sW + ((k >> 1) + hf) * 128;  // pair (kb, kb+1)
#pragma unroll
      for (int c = 0; c < 4; ++c) {
        v2f b = *(const v2f*)(wrow + (c * 16 + r) * 2);  // one ds_load_b64
        acc[c] = wmma4(a, b, acc[c]);
      }
    }
  }

  // relu + store: C layout VGPR i -> row i (+8 for upper half), col c*16+r
#pragma unroll
  for (int c = 0; c < 4; ++c) {
#pragma unroll
    for (int i = 0; i < 8; ++i) {
      int m = rowBase + i + 8 * hf;
      if (m < n) h[(size_t)m * HIDDEN + c * 16 + r] = fmaxf(acc[c][i], 0.0f);
    }
  }
}

// ---------------------------------------------------------------------------
// Fused SAGE layer:
//   out = (agg/deg) @ Wl + bl + h @ Wr
//   out = relu(out * rsqrt(max(sum(out^2), 1e-24)))
//   h'  = layernorm(out + h, g, b)
// One wave = 16 nodes; both 64x64 GEMMs fused in WMMA accumulators; weights
// staged pair-interleaved in LDS (32 KB); row ops via LDS tile + shfl folds.
// ---------------------------------------------------------------------------
__global__ __launch_bounds__(128) void k_layer(const float* __restrict__ hprev,
                                               const float* __restrict__ agg,
                                               const float* __restrict__ deg,
                                               const float* __restrict__ Wl,
                                               const float* __restrict__ bl,
                                               const float* __restrict__ Wr,
                                               const float* __restrict__ g,
                                               const float* __restrict__ bt,
                                               float* __restrict__ hnext, int n) {
  __shared__ float sWl[HIDDEN * HIDDEN];   // 16 KB (pair-interleaved)
  __shared__ float sWr[HIDDEN * HIDDEN];   // 16 KB (pair-interleaved)
  __shared__ float sOut[4 * 16 * HIDDEN];  // 16 KB: per-wave 16x64 tiles

  stage_pairs(sWl, Wl, HIDDEN, threadIdx.x, 128);
  stage_pairs(sWr, Wr, HIDDEN, threadIdx.x, 128);
  __syncthreads();

  const int wave = threadIdx.x >> 5;
  const int lane = threadIdx.x & 31;
  const int hf = lane >> 4;
  const int r = lane & 15;
  const int rowBase = blockIdx.x * 64 + wave * 16;
  const int row = rowBase + r;
  const int rowc = row < n ? row : n - 1;
  const float* ar = agg + (size_t)rowc * HIDDEN;
  const float* hr = hprev + (size_t)rowc * HIDDEN;
  const float invd = 1.0f / fmaxf(deg[rowc], 1.0f);

  v8f acc[4];
#pragma unroll
  for (int c = 0; c < 4; ++c) {
    float b = bl[c * 16 + r];
#pragma unroll
    for (int i = 0; i < 8; ++i) acc[c][i] = b;
  }

#pragma unroll 4
  for (int k = 0; k < HIDDEN; k += 4) {
    v2f a1 = *(const v2f*)(ar + k + 2 * hf);
    a1.x *= invd;
    a1.y *= invd;
    v2f a2 = *(const v2f*)(hr + k + 2 * hf);
    const float* wl = sWl + ((k >> 1) + hf) * 128;
    const float* wr = sWr + ((k >> 1) + hf) * 128;
#pragma unroll
    for (int c = 0; c < 4; ++c) {
      v2f b1 = *(const v2f*)(wl + (c * 16 + r) * 2);  // single ds_load_b64
      acc[c] = wmma4(a1, b1, acc[c]);
      v2f b2 = *(const v2f*)(wr + (c * 16 + r) * 2);
      acc[c] = wmma4(a2, b2, acc[c]);
    }
  }

  // dump C tiles to LDS so each lane can see a contiguous half-row
  float* so = sOut + wave * 16 * HIDDEN;
#pragma unroll
  for (int c = 0; c < 4; ++c)
#pragma unroll
    for (int i = 0; i < 8; ++i)
      so[(i + 8 * hf) * HIDDEN + c * 16 + r] = acc[c][i];
  __syncthreads();

  // row ops: lane handles row r, columns [hf*32, hf*32+32)
  const int node = rowBase + r;
  const bool valid = node < n;
  const int nodec = valid ? node : n - 1;
  const v4f* rp = (const v4f*)(so + r * HIDDEN + hf * 32);
  v4f vv[8];
  float ss = 0.0f;
#pragma unroll
  for (int j = 0; j < 8; ++j) {
    vv[j] = rp[j];
    ss += vv[j].x * vv[j].x + vv[j].y * vv[j].y + vv[j].z * vv[j].z + vv[j].w * vv[j].w;
  }
  ss += __shfl_xor(ss, 16);
  const float l2 = rsqrtf(fmaxf(ss, 1e-24f));

  const v4f* hres = (const v4f*)(hprev + (size_t)nodec * HIDDEN + hf * 32);
  float s = 0.0f, q = 0.0f;
#pragma unroll
  for (int j = 0; j < 8; ++j) {
    v4f t = vv[j] * l2;
    t.x = fmaxf(t.x, 0.0f);
    t.y = fmaxf(t.y, 0.0f);
    t.z = fmaxf(t.z, 0.0f);
    t.w = fmaxf(t.w, 0.0f);
    t += hres[j];
    vv[j] = t;
    s += t.x + t.y + t.z + t.w;
    q += t.x * t.x + t.y * t.y + t.z * t.z + t.w * t.w;
  }
  s += __shfl_xor(s, 16);
  q += __shfl_xor(q, 16);
  const float mean = s * (1.0f / 64.0f);
  const float var = q * (1.0f / 64.0f) - mean * mean;
  const float sc = rsqrtf(var + 1e-5f);

  const v4f* gp = (const v4f*)(g + hf * 32);
  const v4f* bp = (const v4f*)(bt + hf * 32);
  if (valid) {
    v4f* op = (v4f*)(hnext + (size_t)node * HIDDEN + hf * 32);
#pragma unroll
    for (int j = 0; j < 8; ++j) op[j] = (vv[j] - mean) * sc * gp[j] + bp[j];
  }
}

// ---------------------------------------------------------------------------
// Final layernorm: one wave per row, 2 elems per lane, shfl_xor reduction.
// ---------------------------------------------------------------------------
__global__ __launch_bounds__(256) void k_finalln(const float* __restrict__ h,
                                                 const float* __restrict__ g,
                                                 const float* __restrict__ b,
                                                 float* __restrict__ out, int n) {
  int t = blockIdx.x * blockDim.x + threadIdx.x;
  int row = t >> 5;
  int lane = t & 31;
  if (row >= n) return;
  v2f v = *(const v2f*)(h + (size_t)row * HIDDEN + lane * 2);
  float s = v.x + v.y;
  float q = v.x * v.x + v.y * v.y;
#pragma unroll
  for (int m = 16; m >= 1; m >>= 1) {
    s += __shfl_xor(s, m);
    q += __shfl_xor(q, m);
  }
  float mean = s * (1.0f / 64.0f);
  float var = q * (1.0f / 64.0f) - mean * mean;
  float sc = rsqrtf(var + 1e-5f);
  v2f g2 = *(const v2f*)(g + lane * 2);
  v2f b2 = *(const v2f*)(b + lane * 2);
  v2f o;
  o.x = (v.x - mean) * sc * g2.x + b2.x;
  o.y = (v.y - mean) * sc * g2.y + b2.y;
  *(v2f*)(out + (size_t)row * HIDDEN + lane * 2) = o;
}

extern "C" void kernel_launch(void* const* d_in, const int* in_sizes, int n_in,
                              void* d_out, int out_size, void* d_ws, size_t ws_size,
                              hipStream_t stream) {
  const float* x   = (const float*)d_in[0];
  const int* ei    = (const int*)d_in[1];
  const float* Win = (const float*)d_in[2];
  const float* bin = (const float*)d_in[3];
  const float* Wl  = (const float*)d_in[4];
  const float* bl  = (const float*)d_in[5];
  const float* Wr  = (const float*)d_in[6];
  const float* lng = (const float*)d_in[7];
  const float* lnb = (const float*)d_in[8];
  const float* og  = (const float*)d_in[9];
  const float* ob  = (const float*)d_in[10];

  const int N = in_sizes[0] / DNODE;
  const int E = in_sizes[1] / 2;
  const int LAYERS = in_sizes[4] / (HIDDEN * HIDDEN);
  const int* src = ei;
  const int* dst = ei + E;

  // workspace: h0, h1, agg (N*64 floats each) + deg (N floats) ~ 77 MB
  float* h0  = (float*)d_ws;
  float* h1  = h0 + (size_t)N * HIDDEN;
  float* agg = h1 + (size_t)N * HIDDEN;
  float* deg = agg + (size_t)N * HIDDEN;

  // degrees
  k_zero<<<1024, 256, 0, stream>>>(deg, (size_t)N);
  k_deg<<<(E + 255) / 256, 256, 0, stream>>>(dst, deg, E);

  // input projection (fp32 WMMA)
  k_inproj<<<(N + 63) / 64, 128, 0, stream>>>(x, Win, bin, h0, N);

  float* cur = h0;
  float* nxt = h1;
  for (int l = 0; l < LAYERS; ++l) {
    k_zero<<<2048, 256, 0, stream>>>(agg, (size_t)N * HIDDEN);
    k_scatter<<<(int)(((long long)E * 32 + 255) / 256), 256, 0, stream>>>(src, dst, cur, agg, E);
    k_layer<<<(N + 63) / 64, 128, 0, stream>>>(
        cur, agg, deg,
        Wl + (size_t)l * HIDDEN * HIDDEN, bl + (size_t)l * HIDDEN,
        Wr + (size_t)l * HIDDEN * HIDDEN,
        lng + (size_t)l * HIDDEN, lnb + (size_t)l * HIDDEN, nxt, N);
    float* t = cur; cur = nxt; nxt = t;
  }

  k_finalln<<<(int)(((long long)N * 32 + 255) / 256), 256, 0, stream>>>(cur, og, ob, (float*)d_out, N);
}